// HypAgg_5205500363351
// MI455X (gfx1250) — compile-verified
//
#include <hip/hip_runtime.h>
#include <stdint.h>

// HGCN hyperbolic aggregation for MI455X (gfx1250, wave32).
//   out = proj(expmap0(spmm(adj, logmap0(x))))
// Memory-bound: x (51MB) is L2-resident (192MB L2); the SpMM gather is an
// L2-bandwidth problem and the scatter is f32-atomic bound. Phase 2 uses the
// CDNA5 async-to-LDS path (global_load_async_to_lds_b128 + s_wait_asynccnt)
// with a 4-deep per-wave pipeline of 512B row gathers, and chunk-of-32
// coalesced edge-metadata loads broadcast via shuffles.

#define DIMS 128
#define WAVES_PER_BLOCK 8
#define BLOCK 256
#define DEPTH 4   // async pipeline depth (power of two)

// ---------------------------------------------------------------- wave reduce
__device__ __forceinline__ float wave_sum32(float v) {
    v += __shfl_xor(v, 16, 32);
    v += __shfl_xor(v, 8, 32);
    v += __shfl_xor(v, 4, 32);
    v += __shfl_xor(v, 2, 32);
    v += __shfl_xor(v, 1, 32);
    return v;
}

// -------------------------------------------- CDNA5 async gather primitives
// One wave gathers a full 512B row: 32 lanes x b128 (16B each) into LDS.
// saddr form (GVS): addr = SGPR64 base + VGPR32 byte offset.
#define ASYNC_GATHER_ROW(srcbase, ldsaddr, byteoff)                          \
    asm volatile("global_load_async_to_lds_b128 %0, %1, %2"                  \
                 :: "v"(ldsaddr), "v"(byteoff),                              \
                    "s"((unsigned long long)(uintptr_t)(srcbase))            \
                 : "memory")

__device__ __forceinline__ void wait_async_le(int pend) {
    // pend is wave-uniform in [0, DEPTH-1]
    if (pend >= 3)      { asm volatile("s_wait_asynccnt 3" ::: "memory"); }
    else if (pend == 2) { asm volatile("s_wait_asynccnt 2" ::: "memory"); }
    else if (pend == 1) { asm volatile("s_wait_asynccnt 1" ::: "memory"); }
    else                { asm volatile("s_wait_asynccnt 0" ::: "memory"); }
}

// ---------------------------------------------------------------- phase 1
// scale[n] = artanh(clip(max(||x_n||,1e-15)))/max(||x_n||,1e-15)   (c = 1)
__global__ __launch_bounds__(BLOCK) void hyp_logmap_scale(
        const float* __restrict__ x, float* __restrict__ scale, int n) {
    const int wv   = threadIdx.x >> 5;
    const int lane = threadIdx.x & 31;
    const int wave = blockIdx.x * WAVES_PER_BLOCK + wv;
    const int nw   = gridDim.x * WAVES_PER_BLOCK;
    for (int row = wave; row < n; row += nw) {
        float4 q = ((const float4*)(x + (size_t)row * DIMS))[lane];
        float ss = wave_sum32(q.x * q.x + q.y * q.y + q.z * q.z + q.w * q.w);
        float pn = fmaxf(sqrtf(ss), 1e-15f);
        float z  = fminf(pn, 1.0f - 1e-5f);               // artanh clamp (x >= 0 here)
        float a  = 0.5f * (log1pf(z) - log1pf(-z));       // artanh(z)
        if (lane == 0) scale[row] = a / pn;
    }
}

// ---------------------------------------------------------------- phase 2
// For each edge e: out[rows[e]] += vals[e] * scale[cols[e]] * x[cols[e]]
// One wave per chunk of 32 edges; lane i owns edge base+i's metadata
// (coalesced loads), per-edge values broadcast by shuffle; 4-deep async
// double... quad-buffered gather of the 512B source row into LDS.
__global__ __launch_bounds__(BLOCK) void hyp_spmm_scatter(
        const float* __restrict__ x, const int* __restrict__ rows,
        const int* __restrict__ cols, const float* __restrict__ vals,
        const float* __restrict__ scale, float* __restrict__ out, int e_total) {
    __shared__ __align__(16) float lds[WAVES_PER_BLOCK][DEPTH][DIMS];  // 16KB/WG

    const int wv   = threadIdx.x >> 5;
    const int lane = threadIdx.x & 31;
    const int wave = blockIdx.x * WAVES_PER_BLOCK + wv;
    const int nw   = gridDim.x * WAVES_PER_BLOCK;

    // LDS byte addresses for this lane's 16B slot in each buffer.
    // (generic shared pointer low 32 bits == workgroup-relative LDS offset)
    uint32_t ldsa[DEPTH];
#pragma unroll
    for (int b = 0; b < DEPTH; ++b)
        ldsa[b] = (uint32_t)(uintptr_t)&lds[wv][b][lane * 4];
    const uint32_t boff = (uint32_t)(lane * 16);

    for (int base = wave * 32; base < e_total; base += nw * 32) {
        const int cnt = min(32, e_total - base);
        const int idx = base + (lane < cnt ? lane : 0);
        const int   r_l = rows[idx];
        const int   c_l = cols[idx];
        const float w_l = vals[idx] * scale[c_l];   // fused edge weight

        // Prologue: fill the pipeline.
        int issued = (cnt < DEPTH) ? cnt : DEPTH;
#pragma unroll
        for (int k = 0; k < DEPTH; ++k) {
            if (k < issued) {
                const int ck = __shfl(c_l, k, 32);
                ASYNC_GATHER_ROW(x + (size_t)(unsigned)ck * DIMS, ldsa[k], boff);
            }
        }

        for (int j = 0; j < cnt; ++j) {
            wait_async_le(issued - j - 1);          // load j complete (in-order)
            const float w = __shfl(w_l, j, 32);
            const int   r = __shfl(r_l, j, 32);
            float4 q = *(const float4*)&lds[wv][j & (DEPTH - 1)][lane * 4];

            float* ob = out + (size_t)(unsigned)r * DIMS + lane * 4;
            (void)__hip_atomic_fetch_add(ob + 0, w * q.x, __ATOMIC_RELAXED, __HIP_MEMORY_SCOPE_AGENT);
            (void)__hip_atomic_fetch_add(ob + 1, w * q.y, __ATOMIC_RELAXED, __HIP_MEMORY_SCOPE_AGENT);
            (void)__hip_atomic_fetch_add(ob + 2, w * q.z, __ATOMIC_RELAXED, __HIP_MEMORY_SCOPE_AGENT);
            (void)__hip_atomic_fetch_add(ob + 3, w * q.w, __ATOMIC_RELAXED, __HIP_MEMORY_SCOPE_AGENT);

            // Refill: buffer j&3 was just consumed; safe to reissue when
            // issued == j + DEPTH (ds_load data already waited on above).
            if (issued < cnt) {
                const int ci = __shfl(c_l, issued, 32);
                ASYNC_GATHER_ROW(x + (size_t)(unsigned)ci * DIMS,
                                 ldsa[issued & (DEPTH - 1)], boff);
                ++issued;
            }
        }
        // pend reached 0 at j == cnt-1: pipeline fully drained per chunk.
    }
}

// ---------------------------------------------------------------- phase 3
// In-place: y = tanh(||u||) * u / ||u|| ; proj: ||y|| == tanh(||u||)
__global__ __launch_bounds__(BLOCK) void hyp_expmap_proj(
        float* __restrict__ out, int n) {
    const int wv   = threadIdx.x >> 5;
    const int lane = threadIdx.x & 31;
    const int wave = blockIdx.x * WAVES_PER_BLOCK + wv;
    const int nw   = gridDim.x * WAVES_PER_BLOCK;
    const float maxnorm = 1.0f - 4e-3f;
    for (int row = wave; row < n; row += nw) {
        float4* p = (float4*)(out + (size_t)row * DIMS) + lane;
        float4 q = *p;
        float ss = wave_sum32(q.x * q.x + q.y * q.y + q.z * q.z + q.w * q.w);
        float un = fmaxf(sqrtf(ss), 1e-15f);
        float t  = tanhf(un);
        float f  = t / un;                       // expmap0 scale
        float ny = fmaxf(t, 1e-15f);             // ||y||
        if (ny > maxnorm) f *= maxnorm / ny;     // proj
        q.x *= f; q.y *= f; q.z *= f; q.w *= f;
        *p = q;
    }
}

// ---------------------------------------------------------------- launcher
extern "C" void kernel_launch(void* const* d_in, const int* in_sizes, int n_in,
                              void* d_out, int out_size, void* d_ws, size_t ws_size,
                              hipStream_t stream) {
    const float* x    = (const float*)d_in[0];
    const int*   rows = (const int*)d_in[1];
    const int*   cols = (const int*)d_in[2];
    const float* vals = (const float*)d_in[3];

    const int N = in_sizes[0] / DIMS;   // 100000
    const int E = in_sizes[1];          // 1600000

    float* out   = (float*)d_out;
    float* scale = (float*)d_ws;        // N floats (400KB) of scratch

    // d_out is the SpMM accumulator: zero it (capture-safe stream op).
    hipMemsetAsync(d_out, 0, (size_t)out_size * sizeof(float), stream);

    const int row_blocks = (N + WAVES_PER_BLOCK - 1) / WAVES_PER_BLOCK;  // 12500
    hyp_logmap_scale<<<row_blocks, BLOCK, 0, stream>>>(x, scale, N);
    // 2048 blocks * 8 waves = 16384 waves; 50000 chunks of 32 edges -> ~3/wave
    hyp_spmm_scatter<<<2048, BLOCK, 0, stream>>>(x, rows, cols, vals, scale, out, E);
    hyp_expmap_proj<<<row_blocks, BLOCK, 0, stream>>>(out, N);
}